// TopK_23012434772336
// MI455X (gfx1250) — compile-verified
//
#include <hip/hip_runtime.h>
#include <stdint.h>

// ---------------------------------------------------------------------------
// Top-K(64)-of-256 per row with ordered output, gather + tanh epilogue.
// One wave32 per row. Row staged in LDS via CDNA5 async global->LDS DMA.
// Selection: all-descending bitonic top-64 (sort four 64-blocks, mirror-max
// prune 256->128->64 with bitonic cleans) on 32-bit self-indexing keys.
// Cross-lane exchanges within 8-lane groups use DPP8 (pure VALU, 3 ops/CE);
// only the two prune passes (lane-xor 15 / 23) use ds_swizzle.
// ---------------------------------------------------------------------------

#define F_DIM 256
#define KTOP 64
#define ROWS_PER_BLOCK 8

__device__ __forceinline__ unsigned umax32(unsigned a, unsigned b) { return a > b ? a : b; }
__device__ __forceinline__ unsigned umin32(unsigned a, unsigned b) { return a < b ? a : b; }

// DPP8 xor-permute within 8-lane groups: result[lane] = src[lane ^ X], X in [1,7]
template <int X>
__device__ __forceinline__ unsigned dpp8_xor(unsigned v) {
  constexpr int sel = (((0 ^ X) & 7) << 0)  | (((1 ^ X) & 7) << 3)  |
                      (((2 ^ X) & 7) << 6)  | (((3 ^ X) & 7) << 9)  |
                      (((4 ^ X) & 7) << 12) | (((5 ^ X) & 7) << 15) |
                      (((6 ^ X) & 7) << 18) | (((7 ^ X) & 7) << 21);
  return (unsigned)__builtin_amdgcn_mov_dpp8((int)v, sel);
}

// ds_swizzle group-of-32: offset[14:10]=xor, [9:5]=or, [4:0]=and
template <int XM>
__device__ __forceinline__ unsigned swz(unsigned v) {
  return (unsigned)__builtin_amdgcn_ds_swizzle((int)v, (XM << 10) | 0x1F);
}

// in-lane compare-exchange, a keeps max (descending by lower element index)
__device__ __forceinline__ void ce(unsigned &a, unsigned &b) {
  unsigned mx = umax32(a, b);
  unsigned mn = umin32(a, b);
  a = mx; b = mn;
}

// select-based CE vs a cross-lane partner: ((a>p)==keepmax) ? a : p
// lowers to v_cmp + s_xnor (scalar pipe) + v_cndmask -> 3 VALU per element
__device__ __forceinline__ unsigned ce_sel(unsigned a, unsigned p, bool keepmax) {
  bool agt = a > p;
  return (agt == keepmax) ? a : p;
}

// optimal 19-comparator network, sorts k[0..7] descending within the lane
__device__ __forceinline__ void sort8_inlane(unsigned (&k)[8]) {
  ce(k[0], k[1]); ce(k[2], k[3]); ce(k[4], k[5]); ce(k[6], k[7]);
  ce(k[0], k[2]); ce(k[1], k[3]); ce(k[4], k[6]); ce(k[5], k[7]);
  ce(k[1], k[2]); ce(k[5], k[6]); ce(k[0], k[4]); ce(k[3], k[7]);
  ce(k[1], k[5]); ce(k[2], k[6]);
  ce(k[1], k[4]); ce(k[3], k[6]);
  ce(k[2], k[4]); ce(k[3], k[5]);
  ce(k[3], k[4]);
}

// cleanup passes j=4,2,1 (in-register, static direction: lower reg keeps max)
__device__ __forceinline__ void clean_inlane(unsigned (&k)[8]) {
  ce(k[0], k[4]); ce(k[1], k[5]); ce(k[2], k[6]); ce(k[3], k[7]);
  ce(k[0], k[2]); ce(k[1], k[3]); ce(k[4], k[6]); ce(k[5], k[7]);
  ce(k[0], k[1]); ce(k[2], k[3]); ce(k[4], k[5]); ce(k[6], k[7]);
}

// cross-lane xor cleanup pass via DPP8; keepmax precomputed once per pass
template <int X>
__device__ __forceinline__ void cross_clean(unsigned (&k)[8], bool keepmax) {
  unsigned t[8];
#pragma unroll
  for (int r = 0; r < 8; ++r) t[r] = dpp8_xor<X>(k[r]);
#pragma unroll
  for (int r = 0; r < 8; ++r) k[r] = ce_sel(k[r], t[r], keepmax);
}

// mirror pass: partner = lane-xor X (DPP8) + register reversal
template <int X>
__device__ __forceinline__ void mirror_clean(unsigned (&k)[8], bool keepmax) {
  unsigned t[8];
#pragma unroll
  for (int r = 0; r < 8; ++r) t[r] = dpp8_xor<X>(k[7 - r]);
#pragma unroll
  for (int r = 0; r < 8; ++r) k[r] = ce_sel(k[r], t[r], keepmax);
}

// mirror-max prune across >8-lane spans (ds_swizzle); keep max everywhere,
// only the "low half" lanes carry the surviving candidates afterwards
template <int XM>
__device__ __forceinline__ void mirror_prune(unsigned (&k)[8]) {
  unsigned t[8];
#pragma unroll
  for (int r = 0; r < 8; ++r) t[r] = swz<XM>(k[7 - r]);
#pragma unroll
  for (int r = 0; r < 8; ++r) k[r] = umax32(k[r], t[r]);
}

// tanh(x) = 1 - 2/(e^{2x}+1), via v_exp_f32 + v_rcp_f32 (saturates correctly)
__device__ __forceinline__ float fast_tanh(float x) {
  float t = __builtin_amdgcn_exp2f(x * 2.8853900817779268f);  // 2*log2(e)
  return 1.0f - 2.0f * __builtin_amdgcn_rcpf(t + 1.0f);
}

// ---------------------------------------------------------------------------
// prep: w[f] = scorer[f] / ||scorer||  (one 256-thread block)
// ---------------------------------------------------------------------------
__global__ void prep_scorer(const float* __restrict__ scorer, float* __restrict__ w) {
  __shared__ float red[F_DIM];
  int t = threadIdx.x;
  float v = scorer[t];
  red[t] = v * v;
  __syncthreads();
  for (int s = F_DIM / 2; s > 0; s >>= 1) {
    if (t < s) red[t] += red[t + s];
    __syncthreads();
  }
  w[t] = v / sqrtf(red[0]);
}

// ---------------------------------------------------------------------------
// main kernel: one wave per row, 8 rows per 256-thread block
// ---------------------------------------------------------------------------
__global__ __launch_bounds__(256) void topk_tanh_kernel(
    const float* __restrict__ embs, const float* __restrict__ wg,
    float* __restrict__ out, int nrows) {
  __shared__ __align__(16) float smemW[F_DIM];
  __shared__ __align__(16) float smemRows[ROWS_PER_BLOCK * F_DIM];

  const int tid = threadIdx.x;
  const int wv = tid >> 5;
  const int lane = tid & 31;
  const long long row = (long long)blockIdx.x * ROWS_PER_BLOCK + wv;
  const bool active = row < (long long)nrows;

  float* myRow = smemRows + wv * F_DIM;

  // ---- async DMA: row (1024B) global -> LDS, 2 x b128 per lane ----
  if (active) {
    const float* gsrc = embs + (size_t)row * F_DIM;
    unsigned l0 = (unsigned)(uintptr_t)myRow + (unsigned)(lane * 16);
    unsigned l1 = l0 + 512u;
    unsigned long long g0 = (unsigned long long)(uintptr_t)(gsrc + lane * 4);
    unsigned long long g1 = g0 + 512ull;
    asm volatile("global_load_async_to_lds_b128 %0, %1, off"
                 :: "v"(l0), "v"(g0) : "memory");
    asm volatile("global_load_async_to_lds_b128 %0, %1, off"
                 :: "v"(l1), "v"(g1) : "memory");
  }

  // stage normalized scorer weights once per block
  smemW[tid] = wg[tid];
  __syncthreads();
  if (!active) return;  // wave-uniform

  asm volatile("s_wait_asynccnt 0x0" ::: "memory");

  // ---- scores + self-indexing sortable keys (elem e = 8*lane + r) ----
  const float4 e0 = *(const float4*)(myRow + lane * 8);
  const float4 e1 = *(const float4*)(myRow + lane * 8 + 4);
  const float4 w0 = *(const float4*)(smemW + lane * 8);
  const float4 w1 = *(const float4*)(smemW + lane * 8 + 4);

  float sv[8] = {e0.x * w0.x, e0.y * w0.y, e0.z * w0.z, e0.w * w0.w,
                 e1.x * w1.x, e1.y * w1.y, e1.z * w1.z, e1.w * w1.w};
  unsigned k[8];
#pragma unroll
  for (int r = 0; r < 8; ++r) {
    unsigned u = __float_as_uint(sv[r]);
    u ^= ((unsigned)((int)u >> 31)) | 0x80000000u;       // order-preserving map
    k[r] = (u & 0xFFFFFF00u) | (255u - (unsigned)(lane * 8 + r));
  }

  const bool p1 = (lane & 1) == 0;
  const bool p2 = (lane & 2) == 0;
  const bool p4 = (lane & 4) == 0;

  // ---- sort each 64-block (lane-groups of 8) descending ----
  sort8_inlane(k);              // each lane's 8 sorted descending (19 CEs)
  // merge16: mirror (lane-xor1, DPP8) + in-lane cleanup
  mirror_clean<1>(k, p1);
  clean_inlane(k);
  // merge32: mirror (lane-xor3) + xor8 (lane-xor1) + in-lane cleanup
  mirror_clean<3>(k, p2);
  cross_clean<1>(k, p1);
  clean_inlane(k);
  // merge64: mirror (lane-xor7) + xor16 (lane-xor2) + xor8 + in-lane cleanup
  mirror_clean<7>(k, p4);
  cross_clean<2>(k, p2);
  cross_clean<1>(k, p1);
  clean_inlane(k);

  // ---- prune 256->128: mirror-max across block pairs (spans 16 lanes) ----
  mirror_prune<15>(k);          // pair(0,1) survivors -> lanes 0-7; pair(2,3) -> 16-23
  cross_clean<4>(k, p4);        // j=32
  cross_clean<2>(k, p2);        // j=16
  cross_clean<1>(k, p1);        // j=8
  clean_inlane(k);

  // ---- prune 128->64: lanes 0-7 vs reg-mirrored lanes 16-23, clean ----
  mirror_prune<23>(k);
  cross_clean<4>(k, p4);
  cross_clean<2>(k, p2);
  cross_clean<1>(k, p1);
  clean_inlane(k);

  // ---- epilogue: lanes 0-7 hold ranks 8*lane + r, descending ----
  float ov[8];
#pragma unroll
  for (int r = 0; r < 8; ++r) {
    unsigned f = 255u - (k[r] & 255u);   // always in [0,255], safe in all lanes
    float e = myRow[f];
    float s = e * smemW[f];
    ov[r] = e * fast_tanh(s);
  }
  if (lane < 8) {
    float* dst = out + (size_t)row * KTOP + lane * 8;
    *(float4*)(dst)     = make_float4(ov[0], ov[1], ov[2], ov[3]);
    *(float4*)(dst + 4) = make_float4(ov[4], ov[5], ov[6], ov[7]);
  }
}

// ---------------------------------------------------------------------------
extern "C" void kernel_launch(void* const* d_in, const int* in_sizes, int n_in,
                              void* d_out, int out_size, void* d_ws, size_t ws_size,
                              hipStream_t stream) {
  const float* embs = (const float*)d_in[0];
  const float* scorer = (const float*)d_in[1];
  float* out = (float*)d_out;
  float* w = (float*)d_ws;  // 256 floats of scratch

  int nrows = in_sizes[0] / F_DIM;

  prep_scorer<<<1, F_DIM, 0, stream>>>(scorer, w);

  int nblocks = (nrows + ROWS_PER_BLOCK - 1) / ROWS_PER_BLOCK;
  topk_tanh_kernel<<<nblocks, 256, 0, stream>>>(embs, w, out, nrows);
}